// ExponentialMovingAverage_52656299049402
// MI455X (gfx1250) — compile-verified
//
#include <hip/hip_runtime.h>

// EMA scan y_t = (1-a) y_{t-1} + a x_t over [B=16, T=4000, C=512] fp32.
// Blocked-matmul-on-time formulation for CDNA5 WMMA:
//   Y(16t x 16ch) = Mcoef(16x16, lower-tri a*c^(m-k)) * X(16t x 16ch) + c^(m+1)*carry
// Mcoef split into four 16x4 slabs -> 4 chained V_WMMA_F32_16X16X4_F32.
// T chunked (10 x 400) with 64-step warm-up rescan (0.1^64 underflows fp32,
// so chunks are exactly independent) -> 5120 independent waves.
// Chunk index lives in blockIdx.y so loop bounds are scalar: EXEC stays
// all-ones (WMMA requirement) and loop control is SALU-only.
// Coefficient powers 0.1^e computed branchlessly (binary decomposition) so
// the preamble has no divergent loops.

typedef __attribute__((ext_vector_type(2))) float v2f;
typedef __attribute__((ext_vector_type(8))) float v8f;

#define T_DIM     4000
#define C_DIM     512
#define B_DIM     16
#define NCHUNK    10
#define CHUNK_LEN 400          // T_DIM / NCHUNK, multiple of 16
#define WARMUP    64           // multiple of 16; 0.1^64 == 0 in fp32
#define ALPHA_F   0.9f
#define CDECAY    0.1f
#define NCG       ((B_DIM * C_DIM) / 16)   // 512 column groups of 16 channels
#define WAVES_PER_BLOCK 8

// Branchless 0.1^e for e in [0,16]: product over set bits of e.
__device__ __forceinline__ float pow01(int e) {
    const float p1  = CDECAY;
    const float p2  = p1 * p1;
    const float p4  = p2 * p2;
    const float p8  = p4 * p4;
    const float p16 = p8 * p8;
    const int   ec  = (e < 0) ? 0 : e;
    float r = 1.0f;
    r *= (ec & 1)  ? p1  : 1.0f;
    r *= (ec & 2)  ? p2  : 1.0f;
    r *= (ec & 4)  ? p4  : 1.0f;
    r *= (ec & 8)  ? p8  : 1.0f;
    r *= (ec & 16) ? p16 : 1.0f;
    return r;
}

__global__ __launch_bounds__(256) void
ema_wmma_scan_kernel(const float* __restrict__ x, float* __restrict__ y) {
    const int lane = threadIdx.x & 31;
    const int wib  = threadIdx.x >> 5;                     // wave in block
    const int cg    = blockIdx.x * WAVES_PER_BLOCK + wib;  // channel group
    const int chunk = blockIdx.y;                          // time chunk (uniform)

    const int hi = lane >> 4;          // half-wave select
    const int n  = lane & 15;          // B/D column (channel within group)
    const int m  = lane & 15;          // A row (output time within block)

    // flat channel base -> [B, T, C] offsets
    const int    n0   = cg * 16;
    const int    b    = n0 / C_DIM;
    const int    c0   = n0 % C_DIM;
    const size_t base = (size_t)b * T_DIM * C_DIM + (size_t)c0 + (size_t)n;

    // A-matrix slabs: A[g] covers K = 4g..4g+3; per ISA layout lane holds
    // row m, K index = 4g + 2*hi + {0,1}.  coef(m,k) = (k<=m) ? a*c^(m-k) : 0
    v2f A[4];
#pragma unroll
    for (int g = 0; g < 4; ++g) {
        const int k0 = 4 * g + 2 * hi;
        A[g].x = (k0     <= m) ? ALPHA_F * pow01(m - k0)     : 0.0f;
        A[g].y = (k0 + 1 <= m) ? ALPHA_F * pow01(m - k0 - 1) : 0.0f;
    }
    // Carry coefficients: D-row of VGPR r is m_out = r + 8*hi; pc = c^(m_out+1)
    float pc[8];
#pragma unroll
    for (int r = 0; r < 8; ++r) pc[r] = pow01(r + 8 * hi + 1);

    const int t_store = chunk * CHUNK_LEN;                 // scalar
    const int nwarm   = (chunk == 0) ? 0 : (WARMUP / 16);  // scalar trip count

    float carry = 0.0f;

    // pointers for the K-slab rows this lane loads: row k = 4g + 2*hi (+1)
    const float* px = x + base + (size_t)(t_store - nwarm * 16) * C_DIM
                              + (size_t)(2 * hi) * C_DIM;

    // ---- warm-up blocks: compute scan, no stores (re-derives carry) ----
#pragma unroll 1
    for (int it = 0; it < nwarm; ++it) {
        v2f Bm[4];
#pragma unroll
        for (int g = 0; g < 4; ++g) {
            const float* p = px + (size_t)(4 * g) * C_DIM;
            Bm[g].x = p[0];
            Bm[g].y = p[C_DIM];
        }
        px += 16 * C_DIM;
        v8f acc;
#pragma unroll
        for (int r = 0; r < 8; ++r) acc[r] = pc[r] * carry;
        acc = __builtin_amdgcn_wmma_f32_16x16x4_f32(false, A[0], false, Bm[0], (short)0, acc, false, false);
        acc = __builtin_amdgcn_wmma_f32_16x16x4_f32(false, A[1], false, Bm[1], (short)0, acc, false, false);
        acc = __builtin_amdgcn_wmma_f32_16x16x4_f32(false, A[2], false, Bm[2], (short)0, acc, false, false);
        acc = __builtin_amdgcn_wmma_f32_16x16x4_f32(false, A[3], false, Bm[3], (short)0, acc, false, false);
        carry = __shfl(acc[7], 16 + n, 32);   // D row m=15 lives in lanes 16..31, VGPR 7
    }

    // ---- main blocks: compute scan + store (25 iterations, scalar count) ----
    float* py = y + base + (size_t)t_store * C_DIM + (size_t)(8 * hi) * C_DIM;
#pragma unroll 1
    for (int it = 0; it < CHUNK_LEN / 16; ++it) {
        v2f Bm[4];
#pragma unroll
        for (int g = 0; g < 4; ++g) {
            const float* p = px + (size_t)(4 * g) * C_DIM;
            Bm[g].x = p[0];
            Bm[g].y = p[C_DIM];
        }
        px += 16 * C_DIM;
        v8f acc;
#pragma unroll
        for (int r = 0; r < 8; ++r) acc[r] = pc[r] * carry;
        acc = __builtin_amdgcn_wmma_f32_16x16x4_f32(false, A[0], false, Bm[0], (short)0, acc, false, false);
        acc = __builtin_amdgcn_wmma_f32_16x16x4_f32(false, A[1], false, Bm[1], (short)0, acc, false, false);
        acc = __builtin_amdgcn_wmma_f32_16x16x4_f32(false, A[2], false, Bm[2], (short)0, acc, false, false);
        acc = __builtin_amdgcn_wmma_f32_16x16x4_f32(false, A[3], false, Bm[3], (short)0, acc, false, false);
        carry = __shfl(acc[7], 16 + n, 32);

#pragma unroll
        for (int r = 0; r < 8; ++r) py[(size_t)r * C_DIM] = acc[r];
        py += 16 * C_DIM;
    }
}

extern "C" void kernel_launch(void* const* d_in, const int* in_sizes, int n_in,
                              void* d_out, int out_size, void* d_ws, size_t ws_size,
                              hipStream_t stream) {
    (void)in_sizes; (void)n_in; (void)out_size; (void)d_ws; (void)ws_size;
    const float* x = (const float*)d_in[0];
    float*       y = (float*)d_out;

    dim3 grid(NCG / WAVES_PER_BLOCK, NCHUNK);   // (64, 10) blocks
    dim3 block(WAVES_PER_BLOCK * 32);           // 256 threads = 8 waves

    ema_wmma_scan_kernel<<<grid, block, 0, stream>>>(x, y);
}